// GCN_21406117004231
// MI455X (gfx1250) — compile-verified
//
#include <hip/hip_runtime.h>
#include <hip/hip_bf16.h>
#include <math.h>

#define N_NODES 50000
#define N_EDGES 1600000
#define IN_FEATS 256
#define GAMMA 36
#define PAD 48   // padded feature stride: 3 WMMA N-tiles of 16

typedef float v2f __attribute__((ext_vector_type(2)));
typedef float v8f __attribute__((ext_vector_type(8)));

// Interleaved LDS weight layout: (k, c) and (k+1, c) adjacent for even k,
// so a B fragment (two K values, one column) is a single ds_load_b64.
__device__ __forceinline__ int wofs(int k, int c) {
  return (k >> 1) * (2 * PAD) + c * 2 + (k & 1);
}

__device__ __forceinline__ float sigmoidf(float x) {
  return 1.0f / (1.0f + __expf(-x));
}

__global__ void zero_kernel(float* __restrict__ p, long n) {
  long i = (long)blockIdx.x * blockDim.x + threadIdx.x;
  if (i < n) p[i] = 0.0f;
}

__global__ void degree_kernel(const int* __restrict__ src, const int* __restrict__ dst,
                              float* __restrict__ deg_out, float* __restrict__ deg_in) {
  long e = (long)blockIdx.x * blockDim.x + threadIdx.x;
  if (e >= N_EDGES) return;
  atomicAdd(&deg_out[src[e]], 1.0f);
  atomicAdd(&deg_in[dst[e]], 1.0f);
}

// in-place: deg -> clip(deg,1)^-0.5  (applied to both arrays, laid out contiguously)
__global__ void norm_kernel(float* __restrict__ d, long n) {
  long i = (long)blockIdx.x * blockDim.x + threadIdx.x;
  if (i < n) d[i] = rsqrtf(fmaxf(d[i], 1.0f));
}

// Single-block exclusive scan of in-degrees -> CSR row offsets (N_NODES+1 entries).
// 50K elements: each of 1024 threads serially scans a ~49-element chunk, then a
// Hillis-Steele block scan over the chunk sums.
__global__ __launch_bounds__(1024) void scan_kernel(const float* __restrict__ deg,
                                                    int* __restrict__ csr_start) {
  __shared__ int sums[1024];
  const int T = 1024;
  int t = threadIdx.x;
  const int chunk = (N_NODES + T - 1) / T;
  int begin = t * chunk;
  int end = begin + chunk;
  if (end > N_NODES) end = N_NODES;
  int s = 0;
  for (int i = begin; i < end; i++) s += (int)deg[i];
  sums[t] = s;
  __syncthreads();
  for (int off = 1; off < T; off <<= 1) {
    int v = (t >= off) ? sums[t - off] : 0;
    __syncthreads();
    sums[t] += v;
    __syncthreads();
  }
  int run = (t == 0) ? 0 : sums[t - 1];  // exclusive prefix of this chunk
  for (int i = begin; i < end; i++) {
    csr_start[i] = run;
    run += (int)deg[i];
  }
  if (t == T - 1) csr_start[N_NODES] = run;
}

// Scatter edges into CSR-by-dst: csr_src[slot] = src node. 1.6M int atomics, built once.
__global__ void csr_fill_kernel(const int* __restrict__ src, const int* __restrict__ dst,
                                const int* __restrict__ csr_start, int* __restrict__ cursor,
                                int* __restrict__ csr_src) {
  long e = (long)blockIdx.x * blockDim.x + threadIdx.x;
  if (e >= N_EDGES) return;
  int d = dst[e];
  int pos = csr_start[d] + atomicAdd(&cursor[d], 1);
  csr_src[pos] = src[e];
}

// Atomic-free aggregation: out[n, f..f+3] = sum over in-edges of in[src, f..f+3].
// Thread = (node, float4-quad); quads 9..11 write zeros (keeps pad columns clean,
// replaces the separate zero pass). 12 consecutive threads share one CSR range and
// read 144 contiguous bytes per in-edge.
__global__ void gather_agg_kernel(const float* __restrict__ in, const int* __restrict__ csr_start,
                                  const int* __restrict__ csr_src, float* __restrict__ out) {
  long idx = (long)blockIdx.x * blockDim.x + threadIdx.x;  // N_NODES * 12
  if (idx >= (long)N_NODES * 12) return;
  int n = (int)(idx / 12);
  int f = (int)(idx - (long)n * 12) * 4;
  float4 s = make_float4(0.0f, 0.0f, 0.0f, 0.0f);
  if (f < GAMMA) {
    int jb = csr_start[n], je = csr_start[n + 1];
    for (int j = jb; j < je; j++) {
      const float4 v = *(const float4*)(in + (long)csr_src[j] * PAD + f);
      s.x += v.x; s.y += v.y; s.z += v.z; s.w += v.w;
    }
  }
  *(float4*)(out + (long)n * PAD + f) = s;
}

// buf[n,f] = act(buf[n,f] * norm_in[n] + bias[f]) for f < 36; pad cols untouched (stay 0)
__global__ void post_agg_kernel(float* __restrict__ buf, const float* __restrict__ norm_in,
                                const float* __restrict__ bias, int do_relu) {
  long idx = (long)blockIdx.x * blockDim.x + threadIdx.x;  // over N_NODES * 36
  if (idx >= (long)N_NODES * GAMMA) return;
  long n = idx / GAMMA;
  int f = (int)(idx - n * GAMMA);
  float v = buf[n * PAD + f] * norm_in[n] + bias[f];
  if (do_relu) v = fmaxf(v, 0.0f);
  buf[n * PAD + f] = v;
}

// Y[row, 0:48] = (X[row, 0:k_real] * norm[row]) @ Wpad   (Wpad zero-filled to K_PAD x 48)
// Persistent waves: weights staged once per block, each wave loops over 16-row tiles.
template <int K_PAD, int IN_STRIDE>
__global__ __launch_bounds__(128) void node_gemm_kernel(const float* __restrict__ X,
                                                        const float* __restrict__ W,
                                                        int k_real,
                                                        const float* __restrict__ norm,
                                                        float* __restrict__ Y) {
  __shared__ float ldsW[K_PAD * PAD];
  for (int i = threadIdx.x; i < K_PAD * PAD; i += blockDim.x) {
    int k = i / PAD, c = i % PAD;
    ldsW[wofs(k, c)] = (k < k_real && c < GAMMA) ? W[k * GAMMA + c] : 0.0f;
  }
  __syncthreads();

  int wave = threadIdx.x >> 5;
  int lane = threadIdx.x & 31;
  int m = lane & 15;   // row (A) / column (B,D) within tile
  int kh = lane >> 4;  // K-half select per A/B fragment layout
  const long nTiles = (N_NODES + 15) / 16;
  const long wavesTotal = (long)gridDim.x * (blockDim.x >> 5);
  const int iters = (int)((nTiles + wavesTotal - 1) / wavesTotal);

  for (int it = 0; it < iters; it++) {
    long tile = (long)it * wavesTotal + (long)blockIdx.x * (blockDim.x >> 5) + wave;
    long row = tile * 16 + m;
    bool rowOK = row < N_NODES;
    float nrm = rowOK ? norm[row] : 0.0f;
    const float* aPtr = X + (rowOK ? row : 0) * (long)IN_STRIDE;

    v8f acc0 = {}, acc1 = {}, acc2 = {};
    for (int k0 = 0; k0 < K_PAD; k0 += 4) {
      int kk = k0 + 2 * kh;
      float2 av = *(const float2*)(aPtr + kk);
      v2f a;
      a.x = av.x * nrm;
      a.y = av.y * nrm;
      v2f b0 = *(const v2f*)&ldsW[wofs(kk, m)];
      v2f b1 = *(const v2f*)&ldsW[wofs(kk, 16 + m)];
      v2f b2 = *(const v2f*)&ldsW[wofs(kk, 32 + m)];
      acc0 = __builtin_amdgcn_wmma_f32_16x16x4_f32(false, a, false, b0, (short)0, acc0, false, false);
      acc1 = __builtin_amdgcn_wmma_f32_16x16x4_f32(false, a, false, b1, (short)0, acc1, false, false);
      acc2 = __builtin_amdgcn_wmma_f32_16x16x4_f32(false, a, false, b2, (short)0, acc2, false, false);
    }
    // D layout: lanes 0-15 -> N=m, M=j; lanes 16-31 -> N=m, M=j+8
    for (int j = 0; j < 8; j++) {
      long node = tile * 16 + j + 8 * kh;
      if (node < N_NODES) {
        float* y = Y + node * PAD;
        y[m] = acc0[j];
        y[16 + m] = acc1[j];
        y[32 + m] = acc2[j];
      }
    }
  }
}

// Per-wave tile of 16 edges: pair(72) -> sigmoid(36) -> sigmoid(36) -> dot(36) -> out[e].
// Persistent waves (uniform iteration count so in-loop barriers are legal).
// Edge rows staged into wave-private LDS with b128 gathers before the WMMA loops.
__global__ __launch_bounds__(128) void edge_mlp_kernel(
    const float* __restrict__ H, const int* __restrict__ src, const int* __restrict__ dst,
    const float* __restrict__ Wl1, const float* __restrict__ bl1,
    const float* __restrict__ Wl2, const float* __restrict__ bl2,
    const float* __restrict__ Wl3, float* __restrict__ out) {
  __shared__ float ldsW1[72 * PAD];
  __shared__ float ldsW2[PAD * PAD];
  __shared__ float ldsB1[PAD], ldsB2[PAD], ldsW3[PAD];
  __shared__ float pairStage[4][16 * 72];  // per-wave [16 edges][72] concat(src,dst) rows
  __shared__ float stage[4][16 * PAD];     // per-wave hidden activations

  for (int i = threadIdx.x; i < 72 * PAD; i += blockDim.x) {
    int k = i / PAD, c = i % PAD;
    ldsW1[wofs(k, c)] = (c < GAMMA) ? Wl1[k * GAMMA + c] : 0.0f;
  }
  for (int i = threadIdx.x; i < PAD * PAD; i += blockDim.x) {
    int k = i / PAD, c = i % PAD;
    ldsW2[wofs(k, c)] = (k < GAMMA && c < GAMMA) ? Wl2[k * GAMMA + c] : 0.0f;
  }
  if (threadIdx.x < PAD) {
    int t = threadIdx.x;
    ldsB1[t] = (t < GAMMA) ? bl1[t] : 0.0f;
    ldsB2[t] = (t < GAMMA) ? bl2[t] : 0.0f;
    ldsW3[t] = (t < GAMMA) ? Wl3[t] : 0.0f;
  }
  __syncthreads();

  int wave = threadIdx.x >> 5;
  int lane = threadIdx.x & 31;
  int m = lane & 15;
  int kh = lane >> 4;
  float* pr = pairStage[wave];
  float* st = stage[wave];
  const long nTiles = (N_EDGES + 15) / 16;
  const long wavesTotal = (long)gridDim.x * 4;
  const int iters = (int)((nTiles + wavesTotal - 1) / wavesTotal);

  for (int it = 0; it < iters; it++) {
    long etile = (long)it * wavesTotal + (long)blockIdx.x * 4 + wave;

    // stage 16 edges x (36+36) floats with b128 gathers: 9 chunks/lane
    for (int c = 0; c < 9; c++) {
      int chunk = c * 32 + lane;       // 0..287
      int eidx = chunk / 18;           // edge within tile
      int part = chunk - eidx * 18;    // 0..8 = src quads, 9..17 = dst quads
      long ee = etile * 16 + eidx;
      if (ee >= N_EDGES) ee = 0;
      int node = (part < 9) ? src[ee] : dst[ee];
      int fo = (part < 9) ? part * 4 : (part - 9) * 4;
      float4 v = *(const float4*)(H + (long)node * PAD + fo);
      *(float4*)&pr[eidx * 72 + ((part < 9) ? fo : (GAMMA + fo))] = v;
    }
    __syncthreads();

    // layer 1: [16 x 72] @ [72 x 48]
    v8f a0 = {}, a1 = {}, a2 = {};
    for (int k0 = 0; k0 < 72; k0 += 4) {
      int kk = k0 + 2 * kh;
      float2 av = *(const float2*)&pr[m * 72 + kk];
      v2f a;
      a.x = av.x;
      a.y = av.y;
      v2f b0 = *(const v2f*)&ldsW1[wofs(kk, m)];
      v2f b1 = *(const v2f*)&ldsW1[wofs(kk, 16 + m)];
      v2f b2 = *(const v2f*)&ldsW1[wofs(kk, 32 + m)];
      a0 = __builtin_amdgcn_wmma_f32_16x16x4_f32(false, a, false, b0, (short)0, a0, false, false);
      a1 = __builtin_amdgcn_wmma_f32_16x16x4_f32(false, a, false, b1, (short)0, a1, false, false);
      a2 = __builtin_amdgcn_wmma_f32_16x16x4_f32(false, a, false, b2, (short)0, a2, false, false);
    }
    // sigmoid + stage (pad cols forced to 0 so K-padding of next GEMM is exact)
    for (int j = 0; j < 8; j++) {
      int r = j + 8 * kh;
      st[r * PAD + m] = sigmoidf(a0[j] + ldsB1[m]);
      st[r * PAD + 16 + m] = sigmoidf(a1[j] + ldsB1[16 + m]);
      st[r * PAD + 32 + m] = (32 + m < GAMMA) ? sigmoidf(a2[j] + ldsB1[32 + m]) : 0.0f;
    }
    __syncthreads();

    // layer 2: [16 x 48] @ [48 x 48]
    v8f c0 = {}, c1 = {}, c2 = {};
    for (int k0 = 0; k0 < PAD; k0 += 4) {
      int kk = k0 + 2 * kh;
      float2 av = *(const float2*)&st[m * PAD + kk];
      v2f a;
      a.x = av.x;
      a.y = av.y;
      v2f b0 = *(const v2f*)&ldsW2[wofs(kk, m)];
      v2f b1 = *(const v2f*)&ldsW2[wofs(kk, 16 + m)];
      v2f b2 = *(const v2f*)&ldsW2[wofs(kk, 32 + m)];
      c0 = __builtin_amdgcn_wmma_f32_16x16x4_f32(false, a, false, b0, (short)0, c0, false, false);
      c1 = __builtin_amdgcn_wmma_f32_16x16x4_f32(false, a, false, b1, (short)0, c1, false, false);
      c2 = __builtin_amdgcn_wmma_f32_16x16x4_f32(false, a, false, b2, (short)0, c2, false, false);
    }
    __syncthreads();
    for (int j = 0; j < 8; j++) {
      int r = j + 8 * kh;
      st[r * PAD + m] = sigmoidf(c0[j] + ldsB2[m]);
      st[r * PAD + 16 + m] = sigmoidf(c1[j] + ldsB2[16 + m]);
      st[r * PAD + 32 + m] = (32 + m < GAMMA) ? sigmoidf(c2[j] + ldsB2[32 + m]) : 0.0f;
    }
    __syncthreads();

    // final projection: out[e] = hidden2[e,:] . Wl3
    if (lane < 16) {
      long e = etile * 16 + m;
      float s = 0.0f;
      for (int f = 0; f < GAMMA; f++) s += st[m * PAD + f] * ldsW3[f];
      if (e < N_EDGES) out[e] = s;
    }
    __syncthreads();
  }
}

extern "C" void kernel_launch(void* const* d_in, const int* in_sizes, int n_in,
                              void* d_out, int out_size, void* d_ws, size_t ws_size,
                              hipStream_t stream) {
  const float* in_feat = (const float*)d_in[0];
  const int* esrc = (const int*)d_in[1];
  const int* edst = (const int*)d_in[2];
  const float* W1 = (const float*)d_in[3];
  const float* b1 = (const float*)d_in[4];
  const float* W2 = (const float*)d_in[5];
  const float* b2 = (const float*)d_in[6];
  const float* Wl1 = (const float*)d_in[7];
  const float* bl1 = (const float*)d_in[8];
  const float* Wl2 = (const float*)d_in[9];
  const float* bl2 = (const float*)d_in[10];
  const float* Wl3 = (const float*)d_in[11];
  float* out = (float*)d_out;

  // workspace layout: norms (2N f32), 3 padded node-feature buffers (N*48 f32),
  // CSR: start (N+1 i32), cursor (N i32), src list (E i32)  -> ~36 MB total
  float* ws = (float*)d_ws;
  float* norm_out = ws;
  float* norm_in = ws + N_NODES;
  float* bufA = ws + 2 * N_NODES;
  float* bufB = bufA + (size_t)N_NODES * PAD;
  float* bufC = bufB + (size_t)N_NODES * PAD;
  int* csr_start = (int*)(bufC + (size_t)N_NODES * PAD);
  int* cursor = csr_start + N_NODES + 1;
  int* csr_src = cursor + N_NODES;

  const int gemmBlocks = 400;   // persistent waves: 1600 waves, 2 tiles each
  const int mlpBlocks = 2500;   // persistent waves: 10000 waves, 10 tiles each
  const int gatherBlocks = (int)(((long)N_NODES * 12 + 255) / 256);
  const int postBlocks = (int)(((long)N_NODES * GAMMA + 255) / 256);
  const int edgeBlocks = (N_EDGES + 255) / 256;

  // degrees -> CSR offsets (from raw counts) -> scatter edges -> norms (in place)
  zero_kernel<<<(2 * N_NODES + 255) / 256, 256, 0, stream>>>(norm_out, 2L * N_NODES);
  zero_kernel<<<(N_NODES + 255) / 256, 256, 0, stream>>>((float*)cursor, (long)N_NODES);
  degree_kernel<<<edgeBlocks, 256, 0, stream>>>(esrc, edst, norm_out, norm_in);
  scan_kernel<<<1, 1024, 0, stream>>>(norm_in, csr_start);
  csr_fill_kernel<<<edgeBlocks, 256, 0, stream>>>(esrc, edst, csr_start, cursor, csr_src);
  norm_kernel<<<(2 * N_NODES + 255) / 256, 256, 0, stream>>>(norm_out, 2L * N_NODES);

  // GraphConv 1: WMMA GEMM, then atomic-free CSR gather, then norm_in/bias/relu
  node_gemm_kernel<IN_FEATS, IN_FEATS>
      <<<gemmBlocks, 128, 0, stream>>>(in_feat, W1, IN_FEATS, norm_out, bufA);
  gather_agg_kernel<<<gatherBlocks, 256, 0, stream>>>(bufA, csr_start, csr_src, bufB);
  post_agg_kernel<<<postBlocks, 256, 0, stream>>>(bufB, norm_in, b1, 1);

  // GraphConv 2
  node_gemm_kernel<PAD, PAD><<<gemmBlocks, 128, 0, stream>>>(bufB, W2, GAMMA, norm_out, bufC);
  gather_agg_kernel<<<gatherBlocks, 256, 0, stream>>>(bufC, csr_start, csr_src, bufA);
  post_agg_kernel<<<postBlocks, 256, 0, stream>>>(bufA, norm_in, b2, 0);

  // Edge MLP (dominant compute, fp32 WMMA)
  edge_mlp_kernel<<<mlpBlocks, 128, 0, stream>>>(bufA, esrc, edst, Wl1, bl1, Wl2, bl2, Wl3, out);
}